// SeasonalDecompositionLayer_8358006358498
// MI455X (gfx1250) — compile-verified
//
#include <hip/hip_runtime.h>

// Seasonal decomposition (trend / seasonal / residual), fused single pass.
// Shapes hardcoded to the reference: x (4096, 2048) fp32, out (4085, 2048, 3) fp32.

#define T_LEN   2048
#define PERIOD  24
#define HALF    12
#define DROP    11
#define EPT     8      // elements per thread
#define BLOCK   256    // 256 * 8 = 2048 = T_LEN

__global__ __launch_bounds__(BLOCK) void seasonal_decomp_kernel(
    const float* __restrict__ x, float* __restrict__ out)
{
    __shared__ float xs[T_LEN];          // staged input row (8 KB)
    __shared__ float phaseSum[PERIOD];
    __shared__ float smean[PERIOD];

    const int tid = threadIdx.x;
    const int r   = blockIdx.x;          // output row
    const int b   = r + DROP;            // input row (rows 0..10 are dropped)
    const int t0  = tid * EPT;

    if (tid < PERIOD) phaseSum[tid] = 0.0f;

    // ---- CDNA5 async copy: global -> LDS, 32 bytes per lane (2 x b128) ----
    {
        unsigned lbase = (unsigned)(size_t)(&xs[t0]);   // low 32 bits = LDS byte address
        unsigned long long ga =
            (unsigned long long)(const void*)(x + (size_t)b * T_LEN + t0);
        asm volatile(
            "global_load_async_to_lds_b128 %0, %1, off\n\t"
            "global_load_async_to_lds_b128 %0, %1, off offset:16"
            :
            : "v"(lbase), "v"(ga)
            : "memory");
        asm volatile("s_wait_asynccnt 0" ::: "memory");
    }
    __syncthreads();

    // ---- trend: centered moving average with boundary-clipped window ------
    // window(t) = [max(0, t-12), min(T, t+13));  running-sum slide: +x[t+13], -x[t-12]
    float tr[EPT];   // trend
    float dd[EPT];   // detrended = x - trend
    int s = t0 - HALF;     if (s < 0) s = 0;
    int e = t0 + HALF + 1; if (e > T_LEN) e = T_LEN;
    float sum = 0.0f;
    for (int k = s; k < e; ++k) sum += xs[k];
    #pragma unroll
    for (int j = 0; j < EPT; ++j) {
        const int t = t0 + j;
        const float trv = sum / (float)(e - s);
        tr[j] = trv;
        dd[j] = xs[t] - trv;
        const int add = t + HALF + 1;
        const int rem = t - HALF;
        if (add < T_LEN) { sum += xs[add]; e = add + 1; }
        if (rem >= 0)    { sum -= xs[rem]; s = rem + 1; }
    }

    // ---- seasonal: per-phase mean of detrended (LDS f32 atomics) ----------
    #pragma unroll
    for (int j = 0; j < EPT; ++j) {
        atomicAdd(&phaseSum[(t0 + j) % PERIOD], dd[j]);   // ds_add_f32
    }
    __syncthreads();

    if (tid < PERIOD) {
        const int cnt = (T_LEN - tid + PERIOD - 1) / PERIOD;  // 86 for p<8, else 85
        smean[tid] = phaseSum[tid] / (float)cnt;
    }
    __syncthreads();

    // ---- pack (trend, seasonal, residual) and store as 6 x b128 -----------
    float se[EPT], re[EPT];
    #pragma unroll
    for (int j = 0; j < EPT; ++j) {
        se[j] = smean[(t0 + j) % PERIOD];
        re[j] = dd[j] - se[j];          // x - trend - seasonal
    }

    // 24 contiguous floats; base (r*2048 + 8*tid)*3 is a multiple of 8 -> 32B aligned
    float4* op = (float4*)(out + ((size_t)r * T_LEN + t0) * 3);
    #pragma unroll
    for (int q = 0; q < 6; ++q) {
        float v[4];
        #pragma unroll
        for (int k = 0; k < 4; ++k) {
            const int i = 4 * q + k;      // flat index 0..23
            const int j = i / 3;
            const int c = i % 3;
            v[k] = (c == 0) ? tr[j] : ((c == 1) ? se[j] : re[j]);
        }
        op[q] = make_float4(v[0], v[1], v[2], v[3]);
    }
}

extern "C" void kernel_launch(void* const* d_in, const int* in_sizes, int n_in,
                              void* d_out, int out_size, void* d_ws, size_t ws_size,
                              hipStream_t stream) {
    const float* x = (const float*)d_in[0];
    float* out = (float*)d_out;

    const int B    = in_sizes[0] / T_LEN;  // 4096
    const int rows = B - DROP;             // 4085 output rows

    hipLaunchKernelGGL(seasonal_decomp_kernel,
                       dim3(rows), dim3(BLOCK), 0, stream,
                       x, out);
}